// NGCFLayer_66305705115856
// MI455X (gfx1250) — compile-verified
//
#include <hip/hip_runtime.h>
#include <hip/hip_bf16.h>

typedef __attribute__((ext_vector_type(2))) float v2f;
typedef __attribute__((ext_vector_type(8))) float v8f;

#define D 128           // feature dim
#define NTILE 16

// -------------------------------------------------------------------------
// Kernel 1: h = embeds @ W^T  via V_WMMA_F32_16X16X4_F32
// One block (256 thr = 8 waves) handles one 16-row M tile; wave w handles
// the 16-col N tile starting at 16*w.  K=128 -> 32 chained k=4 WMMAs.
//
// f32 WMMA operand layouts (ISA 7.12.2):
//   A 16x4 : VGPR0 = {lanes 0-15: K=0, lanes 16-31: K=2}, VGPR1 = {K=1 | K=3}
//   B 4x16 : VGPR0 = {lanes 0-15: K=0 N=lane, lanes 16-31: K=2 N=lane-16}, etc.
//   C/D 16x16 : VGPR v = {lanes 0-15: M=v, lanes 16-31: M=v+8}, N = lane%16
// -------------------------------------------------------------------------
__global__ __launch_bounds__(256) void gemm_hW_wmma(const float* __restrict__ embeds,
                                                    const float* __restrict__ W,
                                                    float* __restrict__ h,
                                                    int nNodes) {
    const int wave = threadIdx.x >> 5;   // 0..7 -> N tile
    const int lane = threadIdx.x & 31;
    const int half = lane >> 4;          // 0 or 1
    const int l16  = lane & 15;

    const int m0 = blockIdx.x * NTILE;
    const int n0 = wave * NTILE;
    if (m0 >= nNodes) return;            // wave-uniform -> EXEC stays all-1s

    // A: row (m0+l16) of embeds; lane pair-offset 2*half selects K=0/1 vs K=2/3
    const float* arow = embeds + (size_t)(m0 + l16) * D + 2 * half;
    // B[k][n] = W[n][k]  (h = embeds @ W^T); row (n0+l16) of W
    const float* brow = W + (size_t)(n0 + l16) * D + 2 * half;

    v8f c = {};
#pragma unroll
    for (int kk = 0; kk < D; kk += 4) {
        v2f a = *(const v2f*)(arow + kk);   // global_load_b64
        v2f b = *(const v2f*)(brow + kk);   // global_load_b64 (W is L0/L2 hot)
        c = __builtin_amdgcn_wmma_f32_16x16x4_f32(
                /*neg_a=*/false, a, /*neg_b=*/false, b,
                /*c_mod=*/(short)0, c, /*reuse_a=*/false, /*reuse_b=*/false);
    }

    float* orow = h + (size_t)m0 * D + n0;
#pragma unroll
    for (int v = 0; v < 8; ++v) {
        orow[(size_t)(v + 8 * half) * D + l16] = c[v];
    }
}

// -------------------------------------------------------------------------
// Kernel 2: edge scatter.  One wave per edge; lane handles 4 contiguous
// floats of the 128-wide row.  h rows are L2-resident (51.2 MB < 192 MB L2),
// accumulation via hardware GLOBAL_ATOMIC_ADD_F32.
// -------------------------------------------------------------------------
__global__ __launch_bounds__(256) void spmm_scatter(const float* __restrict__ h,
                                                    const int*  __restrict__ rows,
                                                    const int*  __restrict__ cols,
                                                    const float* __restrict__ vals,
                                                    float* __restrict__ out,
                                                    int nEdges) {
    long gid  = (long)blockIdx.x * blockDim.x + threadIdx.x;
    int  e    = (int)(gid >> 5);
    int  lane = (int)(gid & 31);
    if (e >= nEdges) return;

    const int   r = rows[e];
    const int   c = cols[e];
    const float v = vals[e];

    const float4* hv = (const float4*)(h + (size_t)c * D);
    float4 m = hv[lane];

    float* op = out + (size_t)r * D + lane * 4;
    unsafeAtomicAdd(op + 0, v * m.x);
    unsafeAtomicAdd(op + 1, v * m.y);
    unsafeAtomicAdd(op + 2, v * m.z);
    unsafeAtomicAdd(op + 3, v * m.w);
}

// -------------------------------------------------------------------------
// Kernel 3: in-place leaky ReLU (slope 0.2), vectorized float4.
// -------------------------------------------------------------------------
__global__ __launch_bounds__(256) void leaky_relu_inplace(float* __restrict__ out,
                                                          int n4) {
    int i = blockIdx.x * blockDim.x + threadIdx.x;
    if (i >= n4) return;
    float4* p = (float4*)out;
    float4 x = p[i];
    x.x = x.x >= 0.f ? x.x : 0.2f * x.x;
    x.y = x.y >= 0.f ? x.y : 0.2f * x.y;
    x.z = x.z >= 0.f ? x.z : 0.2f * x.z;
    x.w = x.w >= 0.f ? x.w : 0.2f * x.w;
    p[i] = x;
}

extern "C" void kernel_launch(void* const* d_in, const int* in_sizes, int n_in,
                              void* d_out, int out_size, void* d_ws, size_t ws_size,
                              hipStream_t stream) {
    const float* embeds = (const float*)d_in[0];   // [N,128]
    const float* W      = (const float*)d_in[1];   // [128,128]
    const int*   rows   = (const int*)  d_in[2];   // [E]
    const int*   cols   = (const int*)  d_in[3];   // [E]
    const float* vals   = (const float*)d_in[4];   // [E]
    float*       out    = (float*)d_out;           // [N,128]
    float*       h      = (float*)d_ws;            // [N,128] scratch (51.2 MB)

    const int nNodes = in_sizes[0] / D;
    const int nEdges = in_sizes[2];

    // zero accumulator (capturable memset node)
    hipMemsetAsync(out, 0, (size_t)out_size * sizeof(float), stream);

    // 1) h = embeds @ W^T  (WMMA f32)
    int mTiles = (nNodes + NTILE - 1) / NTILE;     // 6250
    gemm_hW_wmma<<<mTiles, 256, 0, stream>>>(embeds, W, h, nNodes);

    // 2) scatter: one wave per edge
    long scatterThreads = (long)nEdges * 32;
    int  scatterBlocks  = (int)((scatterThreads + 255) / 256);
    spmm_scatter<<<scatterBlocks, 256, 0, stream>>>(h, rows, cols, vals, out, nEdges);

    // 3) leaky relu
    int n4 = out_size / 4;
    leaky_relu_inplace<<<(n4 + 255) / 256, 256, 0, stream>>>(out, n4);
}